// RoiPoolingConv_23287312679494
// MI455X (gfx1250) — compile-verified
//
#include <hip/hip_runtime.h>

typedef float v4f __attribute__((ext_vector_type(4)));

#define POOLP     7
#define NROIS     2048
#define IMG_ROWS  128
#define IMG_COLS  128
#define NCH       1024
#define CH4       (NCH / 4)        // float4 lines per pixel = 256 == blockDim
#define LINE_B    (NCH * 4)        // 4096 bytes per (y,x) channel line

// TF2-style half-pixel bilinear coords for one axis:
//   s = (p+0.5)*(size/P) - 0.5, clamped to [0, size-1]
__device__ __forceinline__ void axis_coords(int p, float sizef, int size, int start,
                                            int& i0, int& i1, float& fr) {
    float s = ((float)p + 0.5f) * (sizef / (float)POOLP) - 0.5f;
    float hi = sizef - 1.0f;
    if (hi < 0.0f) hi = 0.0f;
    s = fminf(fmaxf(s, 0.0f), hi);
    float fl = floorf(s);
    int a0 = (int)fl;
    int a1 = a0 + 1;
    int lim = size - 1;
    if (a1 > lim) a1 = lim;
    fr = s - fl;
    i0 = start + a0;
    i1 = start + a1;
}

// Generic pointers to LDS are {aperture[63:32], lds_byte_addr[31:0]} (ISA 10.2):
// truncating gives the wave-relative LDS address the async-load VDST expects.
__device__ __forceinline__ unsigned lds_off(const void* p) {
    return (unsigned)(uintptr_t)p;
}

// GVS-mode async copy: each lane copies 16B from global sbase + byte_off to the LDS address in lds_addr.
// Tracked by ASYNCcnt (CDNA5 §08); per-lane 16B, 4KB per 256-thread block issue.
__device__ __forceinline__ void async_line(unsigned lds_addr, const float* sbase,
                                           unsigned byte_off) {
    asm volatile("global_load_async_to_lds_b128 %0, %1, %2"
                 :: "v"(lds_addr), "v"(byte_off), "s"(sbase)
                 : "memory");
}

// One block per (roi, py): 256 threads x float4 = 1024 channels.
// 2-stage async-to-LDS pipeline over the 7 px cells.
__global__ __launch_bounds__(256) void roi_pool_bilinear(
    const float* __restrict__ img,   // [128,128,1024] NHWC
    const int*   __restrict__ rois,  // [2048,4] = x,y,w,h
    float*       __restrict__ out)   // [2048,7,7,1024]
{
    // double buffer x 4 source lines x 256 float4 slots = 32 KB of 320 KB WGP LDS
    __shared__ v4f smem[2][4][CH4];

    const int blk = blockIdx.x;
    const int roi = blk / POOLP;
    const int py  = blk % POOLP;
    const int t   = threadIdx.x;     // 0..255 -> channels [4t, 4t+3]

    // Uniform per-block ROI fetch (scalarizes to s_load)
    const int rx = rois[roi * 4 + 0];
    const int ry = rois[roi * 4 + 1];
    const int rw = rois[roi * 4 + 2];
    const int rh = rois[roi * 4 + 3];

    int Y0, Y1; float fy;
    axis_coords(py, (float)rh, rh, ry, Y0, Y1, fy);
    const float gy1 = fy, gy0 = 1.0f - fy;

    const int rowY0 = Y0 * IMG_COLS;
    const int rowY1 = Y1 * IMG_COLS;

    // Precompute all 7 x-cell coords (uniform -> SGPRs)
    int X0[POOLP], X1[POOLP]; float FX[POOLP];
    const float wf = (float)rw;
#pragma unroll
    for (int px = 0; px < POOLP; ++px)
        axis_coords(px, wf, rw, rx, X0[px], X1[px], FX[px]);

    v4f* __restrict__ out4 = (v4f*)out;
    const int obase = ((roi * POOLP + py) * POOLP) * CH4 + t;
    const unsigned co = (unsigned)t * 16u;   // this thread's byte offset in a line

    // Issue the 4 async line-copies for cell px into LDS buffer `buf`.
    // Thread t writes only slots [buf][*][t] and later reads only those slots,
    // so per-wave s_wait_asynccnt is sufficient -- no barriers required.
    auto issue = [&](int px, int buf) {
        unsigned a0 = (unsigned)((rowY0 + X0[px]) * LINE_B) + co;
        unsigned a1 = (unsigned)((rowY0 + X1[px]) * LINE_B) + co;
        unsigned a2 = (unsigned)((rowY1 + X0[px]) * LINE_B) + co;
        unsigned a3 = (unsigned)((rowY1 + X1[px]) * LINE_B) + co;
        async_line(lds_off(&smem[buf][0][t]), img, a0);
        async_line(lds_off(&smem[buf][1][t]), img, a1);
        async_line(lds_off(&smem[buf][2][t]), img, a2);
        async_line(lds_off(&smem[buf][3][t]), img, a3);
    };

    issue(0, 0);   // prologue: stage cell 0

#pragma unroll
    for (int px = 0; px < POOLP; ++px) {
        const int cur = px & 1;
        if (px + 1 < POOLP) {
            issue(px + 1, cur ^ 1);                       // prefetch next cell
            asm volatile("s_wait_asynccnt 0x4" ::: "memory"); // current 4 done
        } else {
            asm volatile("s_wait_asynccnt 0x0" ::: "memory");
        }

        // ds_load_b128 x4 from this thread's own LDS slots
        const v4f a = smem[cur][0][t];
        const v4f b = smem[cur][1][t];
        const v4f c = smem[cur][2][t];
        const v4f d = smem[cur][3][t];

        const float gx1 = FX[px], gx0 = 1.0f - gx1;
        v4f r = (a * gx0 + b * gx1) * gy0 + (c * gx0 + d * gx1) * gy1;

        // Streaming output: non-temporal so the 411MB write stream does not
        // evict the 67MB L2-resident image (192MB global L2).
        __builtin_nontemporal_store(r, &out4[obase + px * CH4]);
    }
}

extern "C" void kernel_launch(void* const* d_in, const int* in_sizes, int n_in,
                              void* d_out, int out_size, void* d_ws, size_t ws_size,
                              hipStream_t stream) {
    const float* img  = (const float*)d_in[0];
    const int*   rois = (const int*)d_in[1];
    // d_in[2] is pool_size (==7), baked in as a compile-time constant.
    float* out = (float*)d_out;

    dim3 grid(NROIS * POOLP);   // 14336 blocks
    dim3 block(256);            // 8 wave32s
    roi_pool_bilinear<<<grid, block, 0, stream>>>(img, rois, out);
}